// Grid4D_73040213836176
// MI455X (gfx1250) — compile-verified
//
#include <hip/hip_runtime.h>
#include <cmath>
#include <cstdint>

// ---------------------------------------------------------------------------
// Grid4D hash-grid encode (Instant-NGP / HexPlane style) for MI455X (gfx1250).
// Gather-bound: ~117M 8-byte table gathers against a ~49MB active table set
// (L2-resident on 192MB L2); ~6 GFLOP of fp32 interp math (WMMA inapplicable:
// per-row B matrices). CDNA5-specific paths used:
//   * global_load_async_to_lds_b32 + s_wait_asynccnt  (async coord staging)
//   * LDS-served metadata/coords (ds_load)
//   * non-temporal (TH_STORE_NT) output stores so the write-once 117MB output
//     does not evict the hot table working set from L2.
// ---------------------------------------------------------------------------

#define HASH_T      524288      // 1 << 19
#define NJOB        112         // 16 canon + 3*32 deform levels
#define PTS_PER_BLK 16
#define BLK         128

typedef float v2f __attribute__((ext_vector_type(2)));

struct Params {
  // meta[0..15]  : canonical (xyz) levels   {rx, ry, rz, dense}
  // meta[16..47] : deform levels (shared by xyt/yzt/xzt)
  int4 meta[48];
};

__global__ __launch_bounds__(BLK) void grid4d_kernel(
    const float* __restrict__ xyzt,
    const float* __restrict__ tab_xyz,
    const float* __restrict__ tab_xyt,
    const float* __restrict__ tab_yzt,
    const float* __restrict__ tab_xzt,
    float* __restrict__ out,
    Params p, int N)
{
  __shared__ float s_coord[PTS_PER_BLK * 4];
  __shared__ int4  s_meta[48];

  const int t  = threadIdx.x;
  const int n0 = blockIdx.x * PTS_PER_BLK;

  // --- CDNA5 async copy: stage 16 points x (x,y,z,t) into LDS -------------
  if (t < PTS_PER_BLK * 4 && (n0 * 4 + t) < N * 4) {
    uint32_t lds_off = (uint32_t)(uintptr_t)(&s_coord[t]);
    unsigned long long gaddr =
        (unsigned long long)(uintptr_t)(xyzt + (size_t)n0 * 4 + t);
    asm volatile("global_load_async_to_lds_b32 %0, %1, off"
                 :: "v"(lds_off), "v"(gaddr) : "memory");
  }
  if (t < 48) s_meta[t] = p.meta[t];
  asm volatile("s_wait_asynccnt 0" ::: "memory");
  __syncthreads();

  // --- 14 jobs per thread: job = (point-in-block, level-slot) --------------
  #pragma unroll 2
  for (int it = 0; it < (PTS_PER_BLK * NJOB) / BLK; ++it) {
    int job = it * BLK + t;
    int pi  = job / NJOB;
    int j   = job - pi * NJOB;
    int n   = n0 + pi;
    if (n >= N) continue;

    float px = s_coord[pi * 4 + 0];
    float py = s_coord[pi * 4 + 1];
    float pz = s_coord[pi * 4 + 2];
    float pt = s_coord[pi * 4 + 3];

    float cx, cy, cz;
    const float* tab;
    int lvl, mi;
    size_t obase;
    if (j < 16) {                       // spatial: xyz, canonical levels
      lvl = j;      mi = lvl;      tab = tab_xyz;
      cx = px; cy = py; cz = pz;
      obase = (size_t)n * 32 + 2 * lvl;
    } else if (j < 48) {                // temporal: xyt
      lvl = j - 16; mi = 16 + lvl; tab = tab_xyt;
      cx = px; cy = py; cz = pt;
      obase = (size_t)N * 32 + (size_t)n * 192 + 2 * lvl;
    } else if (j < 80) {                // temporal: yzt
      lvl = j - 48; mi = 16 + lvl; tab = tab_yzt;
      cx = py; cy = pz; cz = pt;
      obase = (size_t)N * 32 + (size_t)n * 192 + 64 + 2 * lvl;
    } else {                            // temporal: xzt
      lvl = j - 80; mi = 16 + lvl; tab = tab_xzt;
      cx = px; cy = pz; cz = pt;
      obase = (size_t)N * 32 + (size_t)n * 192 + 128 + 2 * lvl;
    }

    int4 m = s_meta[mi];
    const int rx = m.x, ry = m.y, rz = m.z, dense = m.w;

    // x01 = clip((x + 1.6) / 3.2, 0, 1)  (fp32, matches reference)
    float ux = fminf(fmaxf((cx + 1.6f) / 3.2f, 0.0f), 1.0f);
    float uy = fminf(fmaxf((cy + 1.6f) / 3.2f, 0.0f), 1.0f);
    float uz = fminf(fmaxf((cz + 1.6f) / 3.2f, 0.0f), 1.0f);

    float posx = ux * (float)(rx - 1);
    float posy = uy * (float)(ry - 1);
    float posz = uz * (float)(rz - 1);

    float fx0 = floorf(posx), fy0 = floorf(posy), fz0 = floorf(posz);
    float fx = posx - fx0, fy = posy - fy0, fz = posz - fz0;
    int ix = (int)fx0, iy = (int)fy0, iz = (int)fz0;

    const float2* lt = (const float2*)tab + (size_t)lvl * HASH_T;

    float acc0 = 0.0f, acc1 = 0.0f;
    #pragma unroll
    for (int c = 0; c < 8; ++c) {
      const int ox = (c >> 2) & 1, oy = (c >> 1) & 1, oz = c & 1;
      int gx = min(ix + ox, rx - 1);
      int gy = min(iy + oy, ry - 1);
      int gz = min(iz + oz, rz - 1);
      uint32_t idx;
      if (dense) {
        idx = (uint32_t)(gx + gy * rx + gz * (rx * ry));
      } else {
        uint32_t h = (uint32_t)gx
                   ^ ((uint32_t)gy * 2654435761u)
                   ^ ((uint32_t)gz * 805459861u);
        idx = h & (uint32_t)(HASH_T - 1);   // == h % T, T is a power of two
      }
      float w = (ox ? fx : 1.0f - fx)
              * (oy ? fy : 1.0f - fy)
              * (oz ? fz : 1.0f - fz);
      float2 v = lt[idx];                   // global_load_b64 gather (RT: cache)
      acc0 += w * v.x;
      acc1 += w * v.y;
    }

    // Write-once output: non-temporal b64 store keeps L2 for the tables.
    v2f r; r.x = acc0; r.y = acc1;
    __builtin_nontemporal_store(r, (v2f*)(out + obase));
  }
}

// ---------------------------------------------------------------------------
// Host side: replicate numpy's resolution schedule exactly (float64 pow/floor,
// per-axis scale), fill Params, launch.
// ---------------------------------------------------------------------------
static void fill_schedule(const double base[3], const double desired[3],
                          int num_levels, int4* meta_out)
{
  for (int l = 0; l < num_levels; ++l) {
    long long r[3];
    for (int d = 0; d < 3; ++d) {
      double scale = pow(desired[d] / base[d], 1.0 / (double)(num_levels - 1));
      r[d] = (long long)floor(base[d] * pow(scale, (double)l));
    }
    long long prod = r[0] * r[1] * r[2];
    meta_out[l] = make_int4((int)r[0], (int)r[1], (int)r[2],
                            (prod <= (long long)HASH_T) ? 1 : 0);
  }
}

extern "C" void kernel_launch(void* const* d_in, const int* in_sizes, int n_in,
                              void* d_out, int out_size, void* d_ws, size_t ws_size,
                              hipStream_t stream)
{
  const float* xyzt = (const float*)d_in[0];
  const float* txyz = (const float*)d_in[1];
  const float* txyt = (const float*)d_in[2];
  const float* tyzt = (const float*)d_in[3];
  const float* txzt = (const float*)d_in[4];
  float* out = (float*)d_out;
  const int N = in_sizes[0] / 4;

  Params p;
  {
    const double base[3] = {16.0, 16.0, 16.0};
    const double des[3]  = {2048.0, 2048.0, 2048.0};
    fill_schedule(base, des, 16, &p.meta[0]);       // CANON_RES
  }
  {
    const double base[3] = {8.0, 8.0, 8.0};
    const double des[3]  = {32.0, 32.0, 16.0};
    fill_schedule(base, des, 32, &p.meta[16]);      // DEFORM_RES
  }

  const int blocks = (N + PTS_PER_BLK - 1) / PTS_PER_BLK;
  grid4d_kernel<<<blocks, BLK, 0, stream>>>(xyzt, txyz, txyt, tyzt, txzt,
                                            out, p, N);
}